// ContrastiveLoss_69449621176854
// MI455X (gfx1250) — compile-verified
//
#include <hip/hip_runtime.h>
#include <hip/hip_bf16.h>

typedef __attribute__((ext_vector_type(2))) float v2f;
typedef __attribute__((ext_vector_type(8))) float v8f;

#define WAVES   8
#define TILE_M  16
#define UNROLL  8        // K columns per body = 4*UNROLL = 32

// One block = 16 rows. Each wave handles a contiguous D/8 column slice and
// accumulates three 16x16 f32 Gram tiles with V_WMMA_F32_16X16X4_F32:
//   a12 = o1_tile * o2_tile^T, a11 = o1*o1^T, a22 = o2*o2^T.
// Diagonals of the accumulators are the per-row dot / sq-norm reductions.
__global__ __launch_bounds__(256) void contrastive_rows_wmma(
    const float* __restrict__ o1, const float* __restrict__ o2,
    const long long* __restrict__ target, float* __restrict__ losses, int D)
{
    const int lane      = threadIdx.x & 31;
    const int wave      = threadIdx.x >> 5;
    const int chunk     = D / WAVES;              // columns per wave (512 for D=4096)
    const int rowInTile = lane & 15;
    const int kHalf     = (lane >> 4) << 1;       // lanes 0-15 -> cols k,k+1; lanes 16-31 -> k+2,k+3

    const size_t row  = (size_t)blockIdx.x * TILE_M + rowInTile;
    const float* p1 = o1 + row * (size_t)D + (size_t)wave * chunk + kHalf;
    const float* p2 = o2 + row * (size_t)D + (size_t)wave * chunk + kHalf;

    v8f a12 = {}; v8f a11 = {}; v8f a22 = {};

    // Row-major 16x4 f32 tile: A-fragment bytes == B-fragment bytes, so each
    // b64 load serves as both operands of the Gram products.
    // Load phase first (16 independent b64 loads in flight), then WMMA phase:
    // maximizes per-wave MLP for this HBM-bound kernel.
    for (int k = 0; k < chunk; k += 4 * UNROLL) {
        v2f x1[UNROLL], x2[UNROLL];
#pragma unroll
        for (int u = 0; u < UNROLL; ++u) {
            x1[u] = *(const v2f*)(p1 + k + 4 * u);
            x2[u] = *(const v2f*)(p2 + k + 4 * u);
        }
#pragma unroll
        for (int u = 0; u < UNROLL; ++u) {
            a12 = __builtin_amdgcn_wmma_f32_16x16x4_f32(false, x1[u], false, x2[u],
                                                        (short)0, a12, false, false);
            a11 = __builtin_amdgcn_wmma_f32_16x16x4_f32(false, x1[u], false, x1[u],
                                                        (short)0, a11, false, false);
            a22 = __builtin_amdgcn_wmma_f32_16x16x4_f32(false, x2[u], false, x2[u],
                                                        (short)0, a22, false, false);
        }
    }

    __shared__ float r12[WAVES][TILE_M];
    __shared__ float r11[WAVES][TILE_M];
    __shared__ float r22[WAVES][TILE_M];

    // Diagonal extraction per the 32-bit C/D layout:
    //   row m (m<8):  VGPR m,   lane m        (N = lane, M = vgpr)
    //   row m (m>=8): VGPR m-8, lane m+16     (N = lane-16, M = 8+vgpr)
    float d12 = 0.f, d11 = 0.f, d22 = 0.f;
    int rowIdx = -1;
#pragma unroll
    for (int r = 0; r < 8; ++r) {
        if (lane == r)        { d12 = a12[r]; d11 = a11[r]; d22 = a22[r]; rowIdx = r;     }
        if (lane == (24 + r)) { d12 = a12[r]; d11 = a11[r]; d22 = a22[r]; rowIdx = 8 + r; }
    }
    if (rowIdx >= 0) {
        r12[wave][rowIdx] = d12;
        r11[wave][rowIdx] = d11;
        r22[wave][rowIdx] = d22;
    }
    __syncthreads();

    // Fixed-order cross-wave sum (deterministic), then per-row loss.
    if (threadIdx.x < TILE_M) {
        float s12 = 0.f, s11 = 0.f, s22 = 0.f;
#pragma unroll
        for (int w = 0; w < WAVES; ++w) {
            s12 += r12[w][threadIdx.x];
            s11 += r11[w][threadIdx.x];
            s22 += r22[w][threadIdx.x];
        }
        const size_t grow = (size_t)blockIdx.x * TILE_M + threadIdx.x;
        const float cosv  = s12 / (sqrtf(s11) * sqrtf(s22));
        const float dist  = 0.5f * (1.0f - cosv);          // 1 - 0.5*(1+cos)
        const float t     = (float)target[grow];
        const float h     = fmaxf(0.0f, 1.0f - sqrtf(dist + 1e-9f));
        losses[grow]      = 0.5f * (t * dist + (1.0f - t) * h * h);
    }
}

// Deterministic fixed-order mean of n floats -> out[0].
__global__ __launch_bounds__(256) void mean_reduce(const float* __restrict__ in,
                                                   float* __restrict__ out, int n)
{
    __shared__ float s[256];
    float acc = 0.f;
    for (int i = threadIdx.x; i < n; i += 256) acc += in[i];
    s[threadIdx.x] = acc;
    __syncthreads();
    for (int off = 128; off > 0; off >>= 1) {
        if ((int)threadIdx.x < off) s[threadIdx.x] += s[threadIdx.x + off];
        __syncthreads();
    }
    if (threadIdx.x == 0) out[0] = s[0] / (float)n;
}

extern "C" void kernel_launch(void* const* d_in, const int* in_sizes, int n_in,
                              void* d_out, int out_size, void* d_ws, size_t ws_size,
                              hipStream_t stream) {
    const float*     o1  = (const float*)d_in[0];
    const float*     o2  = (const float*)d_in[1];
    const long long* tgt = (const long long*)d_in[2];   // jnp int64 targets

    const int B = in_sizes[2];                 // 8192
    const int D = in_sizes[0] / B;             // 4096

    float* losses = (float*)d_ws;              // B floats of scratch

    dim3 grid(B / TILE_M), block(256);
    contrastive_rows_wmma<<<grid, block, 0, stream>>>(o1, o2, tgt, losses, D);
    mean_reduce<<<1, 256, 0, stream>>>(losses, (float*)d_out, B);
}